// TestEBCModel_39582418600476
// MI455X (gfx1250) — compile-verified
//
#include <hip/hip_runtime.h>

#define T_TABLES 4
#define BATCH    8192
#define SEQ_L    50
#define VOCAB_SZ 100000
#define DIM      128
#define NPAD     132          // padded LDS row stride (floats) -> conflict-free
#define TILE_ROWS 128

typedef float v2f __attribute__((ext_vector_type(2)));
typedef float v8f __attribute__((ext_vector_type(8)));

// ---------------------------------------------------------------------------
// Kernel 1: jagged gather + SUM pool. One wave32 per bag; each lane owns a
// float4 (16B) strip of the 512B embedding row -> fully coalesced row reads.
// ---------------------------------------------------------------------------
__global__ __launch_bounds__(256) void ebc_pool_kernel(
    const int* __restrict__ indices, const int* __restrict__ lengths,
    const float* __restrict__ tables, float* __restrict__ out)
{
    const int gtid = blockIdx.x * 256 + threadIdx.x;
    const int s    = gtid >> 5;                 // bag id in [0, T*B)
    const int lane = threadIdx.x & 31;
    if (s >= T_TABLES * BATCH) return;

    const int t   = s >> 13;                    // s / 8192
    const int len = lengths[s];
    const int* idxp = indices + (size_t)s * SEQ_L;
    const float* tab = tables + (size_t)t * VOCAB_SZ * DIM;

    float ax = 0.f, ay = 0.f, az = 0.f, aw = 0.f;
    for (int l = 0; l < len; ++l) {
        const int idx = idxp[l];
        if (l + 1 < len) {
            const int idx2 = idxp[l + 1];
            // global_prefetch_b8: warm next gathered row while we accumulate
            __builtin_prefetch(((const char*)(tab + (size_t)idx2 * DIM)) + lane * 16, 0, 3);
        }
        const float4 v = ((const float4*)(tab + (size_t)idx * DIM))[lane];
        ax += v.x; ay += v.y; az += v.z; aw += v.w;
    }
    float4 r; r.x = ax; r.y = ay; r.z = az; r.w = aw;
    ((float4*)(out + (size_t)s * DIM))[lane] = r;
}

// ---------------------------------------------------------------------------
// Kernel 2: fused 3-layer MLP on x[32768,128] using V_WMMA_F32_16X16X4_F32.
// Each block: 128-row tile, x staged in LDS (ping-pong), W staged per layer.
// ---------------------------------------------------------------------------
__device__ __forceinline__ void stage_tile(const float* __restrict__ src,
                                           float* __restrict__ dst, int tid)
{
    // 128 rows x 32 float4 per row, 256 threads
    for (int i = tid; i < TILE_ROWS * (DIM / 4); i += 256) {
        const int r = i >> 5;
        const int c = i & 31;
        const float4 v = ((const float4*)(src + (size_t)r * DIM))[c];
        *(float4*)(dst + r * NPAD + c * 4) = v;
    }
}

// out[m][n] = bias[n] + sum_k in[m][k] * W[n][k]   (i.e. x @ W.T + b)
__device__ __forceinline__ void mlp_layer(
    const float* __restrict__ bufIn, float* __restrict__ bufOut,
    const float* __restrict__ bufW, const float* __restrict__ bias,
    int wave, int lane, float* __restrict__ gOut, size_t row0)
{
    const int n0   = wave * 16;       // each wave owns a 16-wide column strip
    const int half = lane >> 4;       // lanes 16-31 hold K-pair {2,3} / row M+8
    const int lr   = lane & 15;
    const int kc0  = half * 2;
    const float bn = bias[n0 + lr];   // bias is per output column

    #pragma unroll 1
    for (int m0 = 0; m0 < TILE_ROWS; m0 += 16) {
        v8f acc;
        #pragma unroll
        for (int j = 0; j < 8; ++j) acc[j] = bn;

        const float* aRow = bufIn + (m0 + lr) * NPAD + kc0;  // A: 16x4 f32
        const float* bRow = bufW  + (n0 + lr) * NPAD + kc0;  // B: W[n][k] (col-major of W.T)
        #pragma unroll 8
        for (int k0 = 0; k0 < DIM; k0 += 4) {
            v2f a = *(const v2f*)(aRow + k0);
            v2f b = *(const v2f*)(bRow + k0);
            acc = __builtin_amdgcn_wmma_f32_16x16x4_f32(
                /*neg_a=*/false, a, /*neg_b=*/false, b,
                /*c_mod=*/(short)0, acc, /*reuse_a=*/false, /*reuse_b=*/false);
        }

        // D layout: VGPR j -> row m0 + j + 8*half, col n0 + lr
        if (gOut) {
            #pragma unroll
            for (int j = 0; j < 8; ++j)
                gOut[(row0 + m0 + half * 8 + j) * DIM + n0 + lr] = acc[j];
        } else {
            #pragma unroll
            for (int j = 0; j < 8; ++j)
                bufOut[(m0 + half * 8 + j) * NPAD + n0 + lr] = acc[j];
        }
    }
}

__global__ __launch_bounds__(256) void ebc_mlp_kernel(
    const float* __restrict__ W1, const float* __restrict__ b1,
    const float* __restrict__ W2, const float* __restrict__ b2,
    const float* __restrict__ W3, const float* __restrict__ b3,
    float* __restrict__ xio)
{
    extern __shared__ float lds[];
    float* bufA = lds;                          // 128*132 floats
    float* bufB = lds + TILE_ROWS * NPAD;       // 128*132 floats
    float* bufW = lds + 2 * TILE_ROWS * NPAD;   // 128*132 floats

    const int tid  = threadIdx.x;
    const int wave = tid >> 5;
    const int lane = tid & 31;
    const size_t row0 = (size_t)blockIdx.x * TILE_ROWS;

    stage_tile(xio + row0 * DIM, bufA, tid);
    stage_tile(W1, bufW, tid);
    __syncthreads();
    mlp_layer(bufA, bufB, bufW, b1, wave, lane, nullptr, row0);

    __syncthreads();
    stage_tile(W2, bufW, tid);
    __syncthreads();
    mlp_layer(bufB, bufA, bufW, b2, wave, lane, nullptr, row0);

    __syncthreads();
    stage_tile(W3, bufW, tid);
    __syncthreads();
    mlp_layer(bufA, bufB, bufW, b3, wave, lane, xio, row0);  // final -> global
}

// ---------------------------------------------------------------------------
extern "C" void kernel_launch(void* const* d_in, const int* in_sizes, int n_in,
                              void* d_out, int out_size, void* d_ws, size_t ws_size,
                              hipStream_t stream)
{
    const int*   indices = (const int*)d_in[0];
    const int*   lengths = (const int*)d_in[1];
    const float* tables  = (const float*)d_in[2];
    const float* W1 = (const float*)d_in[3];
    const float* b1 = (const float*)d_in[4];
    const float* W2 = (const float*)d_in[5];
    const float* b2 = (const float*)d_in[6];
    const float* W3 = (const float*)d_in[7];
    const float* b3 = (const float*)d_in[8];
    float* out = (float*)d_out;                 // [32768, 128] f32

    (void)in_sizes; (void)n_in; (void)out_size; (void)d_ws; (void)ws_size;

    const int nBags = T_TABLES * BATCH;         // 32768

    // Stage 1: pooled embeddings written directly into d_out (x buffer).
    const int nThreads1 = nBags * 32;
    ebc_pool_kernel<<<dim3((nThreads1 + 255) / 256), dim3(256), 0, stream>>>(
        indices, lengths, tables, out);

    // Stage 2: fused 3-layer MLP, in-place on d_out (each block reads its
    // 128-row tile fully into LDS before overwriting it).
    const size_t ldsBytes = (size_t)3 * TILE_ROWS * NPAD * sizeof(float); // 198 KB
    ebc_mlp_kernel<<<dim3(nBags / TILE_ROWS), dim3(256), ldsBytes, stream>>>(
        W1, b1, W2, b2, W3, b3, out);
}